// EME_35502199668824
// MI455X (gfx1250) — compile-verified
//
#include <hip/hip_runtime.h>
#include <hip/hip_bf16.h>

// Shapes (fixed by the reference): y_pred = (32, 3, 1024, 1024) fp32, winSize = 8.
// Windows: 96 planes * 128 * 128 = 1,572,864 = 6144 blocks * 256 threads (exact tile).
#define N_PLANES     96
#define ROWSZ        1024           // floats per image row
#define WINS_PER_ROW 128
#define WINS_PER_PLN (128 * 128)    // 16384
#define TOTAL_WINS   (N_PLANES * WINS_PER_PLN)
#define BLK          256
#define NBLOCKS      (TOTAL_WINS / BLK)   // 6144
// final = sum_all * 64 / (1024*1024) / 32 = sum_all / 524288
#define FINAL_SCALE  (1.0f / 524288.0f)

typedef __attribute__((ext_vector_type(2))) float v2f;
typedef __attribute__((ext_vector_type(8))) float v8f;

// Wave32 full-lane sum using V_WMMA_F32_16X16X4_F32 against an all-ones B.
// A layout (16x4 f32): lane m (0..15) holds A[m,0]=VGPR0, A[m,1]=VGPR1;
// lane 16+m holds A[m,2]=VGPR0, A[m,3]=VGPR1.  We place val in VGPR0, 0 in VGPR1,
// so D[m,n] = val[m] + val[m+16].  D VGPR r, lanes 0..15 = D[r,n]; lanes 16..31 =
// D[8+r,n].  Per-lane sum of the 8 D regs gives half-sums; one xor-16 shuffle
// completes the 32-lane total (broadcast to every lane).
__device__ __forceinline__ float wave_sum_wmma(float v) {
    v2f a;  a[0] = v;     a[1] = 0.0f;
    v2f b;  b[0] = 1.0f;  b[1] = 1.0f;   // all-ones B (layout-invariant)
    v8f c = {};
    v8f d = __builtin_amdgcn_wmma_f32_16x16x4_f32(
        /*neg_a=*/false, a, /*neg_b=*/false, b,
        /*c_mod=*/(short)0, c, /*reuse_a=*/false, /*reuse_b=*/false);
    float s = d[0] + d[1] + d[2] + d[3] + d[4] + d[5] + d[6] + d[7];
    s += __shfl_xor(s, 16, 32);
    return s;
}

// Pass 1: one thread per 8x8 window. 16x global_load_b128 (two float4 per row,
// 8 rows). Each wave's paired loads cover a contiguous 1KB span -> full
// cacheline utilization. Min/max chains, masked log, block partial to d_ws.
__global__ __launch_bounds__(BLK)
void eme_window_kernel(const float* __restrict__ y, float* __restrict__ partials) {
    const int gid  = blockIdx.x * BLK + threadIdx.x;          // exact: gid < TOTAL_WINS
    const int p    = gid >> 14;                               // plane (0..95)
    const int win  = gid & (WINS_PER_PLN - 1);
    const int wy   = win >> 7;
    const int wx   = win & (WINS_PER_ROW - 1);

    const float4* base = (const float4*)(y + (size_t)p * (size_t)(ROWSZ * ROWSZ)
                                           + (size_t)wy * (8 * ROWSZ)
                                           + (size_t)wx * 8);
    float mn =  3.402823466e+38f;
    float mx = -3.402823466e+38f;
#pragma unroll
    for (int r = 0; r < 8; ++r) {
        const float4 a = base[r * (ROWSZ / 4)];
        const float4 b = base[r * (ROWSZ / 4) + 1];
        mn = fminf(mn, fminf(fminf(a.x, a.y), fminf(a.z, a.w)));
        mn = fminf(mn, fminf(fminf(b.x, b.y), fminf(b.z, b.w)));
        mx = fmaxf(mx, fmaxf(fmaxf(a.x, a.y), fmaxf(a.z, a.w)));
        mx = fmaxf(mx, fmaxf(fmaxf(b.x, b.y), fmaxf(b.z, b.w)));
    }

    float v = 0.0f;
    if ((mx - mn) != 0.0f)
        v = 20.0f * __logf(mx / (mn + 1.0e-4f));

    // Wave-level sum via WMMA (EXEC is all ones here: no divergence, exact grid).
    const float wsum = wave_sum_wmma(v);

    __shared__ float lds[BLK / 32];
    const int lane = threadIdx.x & 31;
    const int wave = threadIdx.x >> 5;
    if (lane == 0) lds[wave] = wsum;
    __syncthreads();
    if (threadIdx.x == 0) {
        float s = 0.0f;
#pragma unroll
        for (int i = 0; i < BLK / 32; ++i) s += lds[i];
        partials[blockIdx.x] = s;
    }
}

// Pass 2: fold 6144 block partials into the final scaled scalar (deterministic).
__global__ __launch_bounds__(BLK)
void eme_final_kernel(const float* __restrict__ partials, float* __restrict__ out) {
    float s = 0.0f;
    for (int i = threadIdx.x; i < NBLOCKS; i += BLK)
        s += partials[i];
#pragma unroll
    for (int off = 16; off >= 1; off >>= 1)
        s += __shfl_xor(s, off, 32);
    __shared__ float lds[BLK / 32];
    const int lane = threadIdx.x & 31;
    const int wave = threadIdx.x >> 5;
    if (lane == 0) lds[wave] = s;
    __syncthreads();
    if (threadIdx.x == 0) {
        float t = 0.0f;
#pragma unroll
        for (int i = 0; i < BLK / 32; ++i) t += lds[i];
        out[0] = t * FINAL_SCALE;
    }
}

extern "C" void kernel_launch(void* const* d_in, const int* in_sizes, int n_in,
                              void* d_out, int out_size, void* d_ws, size_t ws_size,
                              hipStream_t stream) {
    const float* y     = (const float*)d_in[0];   // (32,3,1024,1024) fp32
    float*       out   = (float*)d_out;           // scalar fp32
    float*       parts = (float*)d_ws;            // 6144 floats = 24 KB scratch

    eme_window_kernel<<<NBLOCKS, BLK, 0, stream>>>(y, parts);
    eme_final_kernel<<<1, BLK, 0, stream>>>(parts, out);
}